// GraphSAGE_498216206707
// MI455X (gfx1250) — compile-verified
//
#include <hip/hip_runtime.h>

typedef float v2f __attribute__((ext_vector_type(2)));
typedef float v8f __attribute__((ext_vector_type(8)));

#define N_NODES 100000
#define N_EDGES 1600000
#define HID     128
#define N_CLS   40

// fp32 WMMA: D(16x16) = A(16x4) * B(4x16) + C, full precision.
__device__ __forceinline__ v8f wmma_f32(v2f a, v2f b, v8f c) {
  return __builtin_amdgcn_wmma_f32_16x16x4_f32(
      /*neg_a=*/false, a, /*neg_b=*/false, b,
      /*c_mod=*/(short)0, c, /*reuse_a=*/false, /*reuse_b=*/false);
}

__device__ __forceinline__ void atom_add(float* p, float v) {
  unsafeAtomicAdd(p, v); // global_atomic_add_f32
}

__global__ void k_zero(float* __restrict__ p, int n) {
  int i = blockIdx.x * blockDim.x + threadIdx.x;
  if (i < n) p[i] = 0.0f;
}

__global__ void k_count(const int* __restrict__ dst, float* __restrict__ cnt, int nE) {
  int e = blockIdx.x * blockDim.x + threadIdx.x;
  if (e < nE) atom_add(&cnt[dst[e]], 1.0f);
}

// One wave per edge; lane l handles 4 channels. Edge index wave-uniform.
__global__ void k_scatter(const float* __restrict__ feat, const int* __restrict__ src,
                          const int* __restrict__ dst, float* __restrict__ agg, int nE) {
  int gid  = blockIdx.x * blockDim.x + threadIdx.x;
  int e    = gid >> 5;
  int lane = gid & 31;
  if (e >= nE) return;
  int s = src[e], d = dst[e];
  float4 v = ((const float4*)(feat + (size_t)s * HID))[lane];
  float* p = agg + (size_t)d * HID + lane * 4;
  atom_add(p + 0, v.x);
  atom_add(p + 1, v.y);
  atom_add(p + 2, v.z);
  atom_add(p + 3, v.w);
}

// h = relu(mean @ Wl + x @ Wr + b), 16 nodes x 128 outs per block (8 waves).
__global__ void __launch_bounds__(256) k_gemm1(
    const float* __restrict__ agg, const float* __restrict__ cnt,
    const float* __restrict__ x,   const float* __restrict__ Wl,
    const float* __restrict__ Wr,  const float* __restrict__ bias,
    float* __restrict__ h) {
  __shared__ float sM[16 * 132];  // stride 132: conflict-free column reads, 16B-aligned rows
  __shared__ float sX[16 * 132];
  const int tid = threadIdx.x;
  const int node0 = blockIdx.x * 16;

  for (int i = tid; i < 16 * 32; i += 256) {   // 32 float4 per row
    int r = i >> 5, c4 = i & 31;
    float4 va = ((const float4*)(agg + (size_t)(node0 + r) * HID))[c4];
    float4 vx = ((const float4*)(x   + (size_t)(node0 + r) * HID))[c4];
    float inv = 1.0f / fmaxf(cnt[node0 + r], 1.0f);
    va.x *= inv; va.y *= inv; va.z *= inv; va.w *= inv;
    ((float4*)(sM + r * 132))[c4] = va;
    ((float4*)(sX + r * 132))[c4] = vx;
  }
  __syncthreads();

  const int lane = tid & 31;
  const int wave = tid >> 5;
  const int half = lane >> 4;          // 0 or 1
  const int koff = 2 * half;           // A/B: lanes 16-31 hold K+2
  const int row  = lane & 15;
  const int col  = wave * 16 + row;

  v8f acc = {0.f, 0.f, 0.f, 0.f, 0.f, 0.f, 0.f, 0.f};
  const float* mrow = sM + row * 132;
  const float* xrow = sX + row * 132;

  #pragma unroll 4
  for (int k = 0; k < HID; k += 4) {
    v2f aM, aX, bL, bR;
    aM[0] = mrow[k + koff];  aM[1] = mrow[k + koff + 1];
    aX[0] = xrow[k + koff];  aX[1] = xrow[k + koff + 1];
    bL[0] = Wl[(k + koff) * HID + col];  bL[1] = Wl[(k + koff + 1) * HID + col];
    bR[0] = Wr[(k + koff) * HID + col];  bR[1] = Wr[(k + koff + 1) * HID + col];
    acc = wmma_f32(aM, bL, acc);
    acc = wmma_f32(aX, bR, acc);
  }

  const float bv = bias[col];
  const int rbase = 8 * half;          // D: lanes 16-31 hold M+8
  #pragma unroll
  for (int i = 0; i < 8; ++i) {
    float v = fmaxf(acc[i] + bv, 0.0f);
    h[(size_t)(node0 + rbase + i) * HID + col] = v;
  }
}

// out = mean2 @ Wl2 + h @ Wr2 + b2 ; N_CLS=40 -> weights zero-padded to 48 in LDS
// so WMMA waves run with full EXEC; only the store masks col < 40.
__global__ void __launch_bounds__(128) k_gemm2(
    const float* __restrict__ agg, const float* __restrict__ cnt,
    const float* __restrict__ h,   const float* __restrict__ Wl,
    const float* __restrict__ Wr,  const float* __restrict__ bias,
    float* __restrict__ out) {
  __shared__ float sM[16 * 132];
  __shared__ float sX[16 * 132];
  __shared__ float sWl[HID * 48];
  __shared__ float sWr[HID * 48];
  const int tid = threadIdx.x;
  const int node0 = blockIdx.x * 16;

  for (int i = tid; i < HID * 48; i += 128) {
    int k = i / 48, c = i - k * 48;
    float vl = 0.0f, vr = 0.0f;
    if (c < N_CLS) { vl = Wl[k * N_CLS + c]; vr = Wr[k * N_CLS + c]; }
    sWl[i] = vl;
    sWr[i] = vr;
  }
  for (int i = tid; i < 16 * 32; i += 128) {
    int r = i >> 5, c4 = i & 31;
    float4 va = ((const float4*)(agg + (size_t)(node0 + r) * HID))[c4];
    float4 vx = ((const float4*)(h   + (size_t)(node0 + r) * HID))[c4];
    float inv = 1.0f / fmaxf(cnt[node0 + r], 1.0f);
    va.x *= inv; va.y *= inv; va.z *= inv; va.w *= inv;
    ((float4*)(sM + r * 132))[c4] = va;
    ((float4*)(sX + r * 132))[c4] = vx;
  }
  __syncthreads();

  const int wave = tid >> 5;
  if (wave >= 3) return;               // whole-wave exit; WMMA waves keep EXEC all-1s
  const int lane = tid & 31;
  const int half = lane >> 4;
  const int koff = 2 * half;
  const int row  = lane & 15;
  const int col  = wave * 16 + row;    // 0..47

  v8f acc = {0.f, 0.f, 0.f, 0.f, 0.f, 0.f, 0.f, 0.f};
  const float* mrow = sM + row * 132;
  const float* xrow = sX + row * 132;

  #pragma unroll 4
  for (int k = 0; k < HID; k += 4) {
    v2f aM, aX, bL, bR;
    aM[0] = mrow[k + koff];  aM[1] = mrow[k + koff + 1];
    aX[0] = xrow[k + koff];  aX[1] = xrow[k + koff + 1];
    bL[0] = sWl[(k + koff) * 48 + col];  bL[1] = sWl[(k + koff + 1) * 48 + col];
    bR[0] = sWr[(k + koff) * 48 + col];  bR[1] = sWr[(k + koff + 1) * 48 + col];
    acc = wmma_f32(aM, bL, acc);
    acc = wmma_f32(aX, bR, acc);
  }

  if (col < N_CLS) {
    const float bv = bias[col];
    const int rbase = 8 * half;
    #pragma unroll
    for (int i = 0; i < 8; ++i) {
      out[(size_t)(node0 + rbase + i) * N_CLS + col] = acc[i] + bv;
    }
  }
}

extern "C" void kernel_launch(void* const* d_in, const int* in_sizes, int n_in,
                              void* d_out, int out_size, void* d_ws, size_t ws_size,
                              hipStream_t stream) {
  const float* x   = (const float*)d_in[0];
  const int*   ei  = (const int*)d_in[1];
  const float* Wl1 = (const float*)d_in[2];
  const float* Wr1 = (const float*)d_in[3];
  const float* b1  = (const float*)d_in[4];
  const float* Wl2 = (const float*)d_in[5];
  const float* Wr2 = (const float*)d_in[6];
  const float* b2  = (const float*)d_in[7];
  const int* src = ei;
  const int* dst = ei + N_EDGES;

  float* agg = (float*)d_ws;                    // N_NODES*HID
  float* cnt = agg + (size_t)N_NODES * HID;     // N_NODES
  float* h   = cnt + N_NODES;                   // N_NODES*HID

  // zero agg + cnt (contiguous)
  {
    int n = N_NODES * HID + N_NODES;
    k_zero<<<(n + 255) / 256, 256, 0, stream>>>(agg, n);
  }
  k_count<<<(N_EDGES + 255) / 256, 256, 0, stream>>>(dst, cnt, N_EDGES);

  const unsigned scatterBlocks = (unsigned)(((long long)N_EDGES * 32 + 255) / 256);
  k_scatter<<<scatterBlocks, 256, 0, stream>>>(x, src, dst, agg, N_EDGES);
  k_gemm1<<<N_NODES / 16, 256, 0, stream>>>(agg, cnt, x, Wl1, Wr1, b1, h);

  {
    int n = N_NODES * HID;
    k_zero<<<(n + 255) / 256, 256, 0, stream>>>(agg, n);
  }
  k_scatter<<<scatterBlocks, 256, 0, stream>>>(h, src, dst, agg, N_EDGES);
  k_gemm2<<<N_NODES / 16, 128, 0, stream>>>(agg, cnt, h, Wl2, Wr2, b2, (float*)d_out);
}